// InitAll_20933670601054
// MI455X (gfx1250) — compile-verified
//
#include <hip/hip_runtime.h>

#define NNODE 6000
#define NELEM 12000
#define NFIX 200
#define NDLOAD 400
#define NDOF (2 * NNODE)          // 12000
#define NPAIR (NELEM / 2)         // 6000 (two elements per wave-WMMA)

typedef float v2f __attribute__((ext_vector_type(2)));
typedef float v8f __attribute__((ext_vector_type(8)));

// ---------------------------------------------------------------------------
// Kernel 1: zero-fill the whole output (stiff 144M floats + resid 12K floats).
// 576 MB of stores -> this IS the roofline (~25 us at 23.3 TB/s). B128 stores.
// ---------------------------------------------------------------------------
__global__ void fem_zero_kernel(float4* __restrict__ out, size_t n4) {
    size_t i = (size_t)blockIdx.x * blockDim.x + threadIdx.x;
    size_t stride = (size_t)gridDim.x * blockDim.x;
    float4 z; z.x = 0.f; z.y = 0.f; z.z = 0.f; z.w = 0.f;
    for (; i < n4; i += stride) out[i] = z;
}

// ---------------------------------------------------------------------------
// Kernel 2: element assembly. One wave processes TWO elements with a single
// V_WMMA_F32_16X16X4_F32:  C = A(16x4) x B(4x16)
//   A rows 0-5  = e0's B^T (6x3, K=3 padded 0), rows 8-13 = e1's B^T
//   B cols 0-5  = e0's M = area*D*B (3x6),      cols 8-13 = e1's M
// Diagonal 6x6 blocks of C are k0, k1 (full f32 precision). Cross blocks are
// ignored. 12 active lanes then scatter 6 atomic f32 adds each (2x36 total).
//
// ISA A layout (16x4 f32, 2 VGPRs): lane L row = L%16; lanes 0-15 hold K=0,1;
// lanes 16-31 hold K=2,3.  B layout (4x16) mirrors: lanes 0-15 K-rows 0,1;
// lanes 16-31 K-rows 2,3; lane column = L%16.
// ---------------------------------------------------------------------------
__global__ void fem_assemble_kernel(const float* __restrict__ xcoord,
                                    const float* __restrict__ ycoord,
                                    const int*   __restrict__ connect,
                                    const float* __restrict__ elem_material,
                                    float* __restrict__ stiff) {
    const int lane = threadIdx.x & 31;
    const int wave = threadIdx.x >> 5;
    const int pair = blockIdx.x * (blockDim.x >> 5) + wave;
    if (pair >= NPAIR) return;                 // wave-uniform guard

    const int n    = lane & 15;                // row/col index in 16-space
    const int half = lane >> 4;                // 0 -> K=0,1 ; 1 -> K=2,3
    const int e    = pair * 2 + (n >> 3);      // n<8 -> elem0, n>=8 -> elem1
    const int d    = n & 7;                    // local dof 0..7 (valid <6)

    // --- element geometry (each lane loads its own element's 3 nodes) ---
    const int na = connect[3 * e + 0];
    const int nb = connect[3 * e + 1];
    const int nc = connect[3 * e + 2];
    const float xa = xcoord[na], xb = xcoord[nb], xc = xcoord[nc];
    const float ya = ycoord[na], yb = ycoord[nb], yc = ycoord[nc];

    const float den_a = (ya - yb) * (xc - xb) - (xa - xb) * (yc - yb);
    const float den_b = (yb - yc) * (xa - xc) - (xb - xc) * (ya - yc);
    const float den_c = (yc - ya) * (xb - xa) - (xc - xa) * (yb - ya);
    const float nax = -(yc - yb) / den_a, nay = (xc - xb) / den_a;
    const float nbx = -(ya - yc) / den_b, nby = (xa - xc) / den_b;
    const float ncx = -(yb - ya) / den_c, ncy = (xb - xa) / den_c;
    const float area = 0.5f * fabsf((xb - xa) * (yc - ya) - (xc - xa) * (yb - ya));

    // --- material matrix entries (D symmetric: [[d1,d2,0],[d2,d1,0],[0,0,d3]]) ---
    const float c_st = 200.0f / ((1.0f + 0.3f) * (1.0f - 2.0f * 0.3f));
    const float c_al = 69.0f  / ((1.0f + 0.33f) * (1.0f - 2.0f * 0.33f));
    const bool  alu  = elem_material[e] > 0.5f;
    const float D1 = alu ? c_al * (1.0f - 0.33f) : c_st * (1.0f - 0.3f);
    const float D2 = alu ? c_al * 0.33f          : c_st * 0.3f;
    const float D3 = alu ? c_al * ((1.0f - 2.0f * 0.33f) * 0.5f)
                         : c_st * ((1.0f - 2.0f * 0.3f)  * 0.5f);

    // --- B[:,d] for local dof d: node = d/2; even d -> (nx,0,ny), odd -> (0,ny,nx)
    const int  node = d >> 1;
    const float nx = (node == 0) ? nax : ((node == 1) ? nbx : ncx);
    const float ny = (node == 0) ? nay : ((node == 1) ? nby : ncy);
    const bool  xdof  = ((d & 1) == 0);
    const bool  valid = (d < 6);

    // B column entries
    const float b0c = xdof ? nx : 0.0f;
    const float b1c = xdof ? 0.0f : ny;
    const float b2c = xdof ? ny : nx;
    // M[:,d] = area * D * B[:,d]
    const float m0 = area * (xdof ? D1 * nx : D2 * ny);
    const float m1 = area * (xdof ? D2 * nx : D1 * ny);
    const float m2 = area * (xdof ? D3 * ny : D3 * nx);

    v2f Aop, Bop;
    if (half == 0) { Aop.x = b0c; Aop.y = b1c; Bop.x = m0; Bop.y = m1; }
    else           { Aop.x = b2c; Aop.y = 0.f; Bop.x = m2; Bop.y = 0.f; }  // K=3 row = 0
    if (!valid)    { Aop.x = 0.f; Aop.y = 0.f; Bop.x = 0.f; Bop.y = 0.f; }

    v8f Cacc = {};
    // D = A(16x4,f32) x B(4x16,f32) + 0 ; full f32 precision
    Cacc = __builtin_amdgcn_wmma_f32_16x16x4_f32(
        /*neg_a=*/false, Aop, /*neg_b=*/false, Bop,
        /*c_mod=*/(short)0, Cacc, /*reuse_a=*/false, /*reuse_b=*/false);

    // --- scatter: lanes 0-5 own e0 columns (rows 0-7 in VGPRs 0-7),
    //              lanes 24-29 own e1 columns (rows 8-15 in VGPRs 0-7) ---
    const bool active = valid && ((half == 0) ? (n < 8) : (n >= 8));
    if (active) {
        int gd0 = 2 * na, gd1 = 2 * na + 1;
        int gd2 = 2 * nb, gd3 = 2 * nb + 1;
        int gd4 = 2 * nc, gd5 = 2 * nc + 1;
        int gd[6] = {gd0, gd1, gd2, gd3, gd4, gd5};
        const int coldof = gd[d];
#pragma unroll
        for (int v = 0; v < 6; ++v) {
            atomicAdd(&stiff[(size_t)gd[v] * NDOF + coldof], Cacc[v]);
        }
    }
}

// ---------------------------------------------------------------------------
// Kernel 3: traction loads into resid (400 entries, 4 atomics each).
// ---------------------------------------------------------------------------
__global__ void fem_traction_kernel(const float* __restrict__ xcoord,
                                    const float* __restrict__ ycoord,
                                    const int*   __restrict__ connect,
                                    const int*   __restrict__ dload_elem,
                                    const int*   __restrict__ dload_face,
                                    const float* __restrict__ dload_tx,
                                    const float* __restrict__ dload_ty,
                                    float* __restrict__ resid) {
    int i = blockIdx.x * blockDim.x + threadIdx.x;
    if (i >= NDLOAD) return;
    const int e = dload_elem[i];
    const int f = dload_face[i];
    const int nxt = (f == 0) ? 1 : ((f == 1) ? 2 : 0);   // pointer = [1,2,0]
    const int a = connect[3 * e + f];
    const int b = connect[3 * e + nxt];
    const float dx = xcoord[a] - xcoord[b];
    const float dy = ycoord[a] - ycoord[b];
    const float hl = 0.5f * sqrtf(dx * dx + dy * dy);
    const float fx = dload_tx[i] * hl;
    const float fy = dload_ty[i] * hl;
    atomicAdd(&resid[2 * a],     fx);
    atomicAdd(&resid[2 * a + 1], fy);
    atomicAdd(&resid[2 * b],     fx);
    atomicAdd(&resid[2 * b + 1], fy);
}

// ---------------------------------------------------------------------------
// Kernel 4: BC row zeroing (one block per fixed dof, idempotent on dupes).
// ---------------------------------------------------------------------------
__global__ void fem_bc_zero_rows_kernel(const int* __restrict__ fix_node,
                                        const int* __restrict__ fix_dof,
                                        float* __restrict__ stiff) {
    const int fi = blockIdx.x;                           // 0..NFIX-1
    const int rw = 2 * fix_node[fi] + fix_dof[fi];
    float* row = stiff + (size_t)rw * NDOF;
    for (int c = threadIdx.x; c < NDOF; c += blockDim.x) row[c] = 0.0f;
}

// ---------------------------------------------------------------------------
// Kernel 5: BC diagonal + resid sets, serial in index order (last-wins for
// duplicate fixes, matching scatter-set semantics). Runs after row zeroing.
// ---------------------------------------------------------------------------
__global__ void fem_bc_set_kernel(const int* __restrict__ fix_node,
                                  const int* __restrict__ fix_dof,
                                  const float* __restrict__ fix_val,
                                  float* __restrict__ stiff,
                                  float* __restrict__ resid) {
    if (blockIdx.x == 0 && threadIdx.x == 0) {
        for (int i = 0; i < NFIX; ++i) {
            const int rw = 2 * fix_node[i] + fix_dof[i];
            stiff[(size_t)rw * NDOF + rw] = 1.0f;
            resid[rw] = fix_val[i];
        }
    }
}

// ---------------------------------------------------------------------------
extern "C" void kernel_launch(void* const* d_in, const int* in_sizes, int n_in,
                              void* d_out, int out_size, void* d_ws, size_t ws_size,
                              hipStream_t stream) {
    (void)in_sizes; (void)n_in; (void)d_ws; (void)ws_size; (void)out_size;

    const float* xcoord        = (const float*)d_in[0];
    const float* ycoord        = (const float*)d_in[1];
    const int*   connect       = (const int*)  d_in[2];
    const float* elem_material = (const float*)d_in[3];
    const int*   dload_elem    = (const int*)  d_in[4];
    const int*   dload_face    = (const int*)  d_in[5];
    const float* dload_tx      = (const float*)d_in[6];
    const float* dload_ty      = (const float*)d_in[7];
    const int*   fix_node      = (const int*)  d_in[8];
    const int*   fix_dof       = (const int*)  d_in[9];
    const float* fix_val       = (const float*)d_in[10];

    float* stiff = (float*)d_out;                       // NDOF*NDOF floats
    float* resid = stiff + (size_t)NDOF * NDOF;         // NDOF floats

    // 1) zero 576 MB output (the roofline cost of this problem)
    const size_t n4 = ((size_t)NDOF * NDOF + NDOF) / 4; // divisible by 4
    fem_zero_kernel<<<4096, 256, 0, stream>>>((float4*)d_out, n4);

    // 2) WMMA element assembly: 8 waves/block, 2 elements/wave -> 750 blocks
    fem_assemble_kernel<<<NPAIR / 8, 256, 0, stream>>>(
        xcoord, ycoord, connect, elem_material, stiff);

    // 3) traction loads
    fem_traction_kernel<<<(NDLOAD + 255) / 256, 256, 0, stream>>>(
        xcoord, ycoord, connect, dload_elem, dload_face, dload_tx, dload_ty, resid);

    // 4) zero fixed rows, then 5) diagonal/resid sets (stream-ordered)
    fem_bc_zero_rows_kernel<<<NFIX, 256, 0, stream>>>(fix_node, fix_dof, stiff);
    fem_bc_set_kernel<<<1, 32, 0, stream>>>(fix_node, fix_dof, fix_val, stiff, resid);
}